// Diffusion_55087250539164
// MI455X (gfx1250) — compile-verified
//
#include <hip/hip_runtime.h>
#include <stdint.h>

// Problem geometry (static in the reference's setup_inputs).
#define H       128
#define W       128
#define PLANE   (H * W)        // 16384 floats per (n,c) plane
#define TILE_H  16             // output rows per block
#define TPP     (H / TILE_H)   // tiles per plane = 8
#define LDS_ROWS 18            // TILE_H + 2 halo rows
#define LDS_W   136            // 4-float left pad + 128 interior + 4 right pad (16B-aligned interior)
#define NTHREADS 256           // 8 wave32

// One diffusion step:
//   out[p, gh, gw] = sum_k ( |w[p,k,gh,gw]| / sum_k|w| ) * xpad[gh+ki-1, gw+kj-1]
// Per-pixel-varying taps => elementwise 9-tap dot, not a GEMM (WMMA inapplicable).
// Bandwidth-bound at ~0.25 flop/byte; weights (151MB) stay L2-resident across steps.
//
// LDS column mapping: image column gw lives at LDS col (gw + 4).
//   left halo (gw=-1)  -> col 3   (always zero)
//   right halo (gw=128)-> col 132 (always zero)
__global__ __launch_bounds__(NTHREADS)
void diffusion_step(const float* __restrict__ xin,
                    const float* __restrict__ wgt,
                    float* __restrict__ xout)
{
    const int tid   = threadIdx.x;
    const int plane = blockIdx.x / TPP;    // n*C + c
    const int tyb   = blockIdx.x % TPP;
    const int row0  = tyb * TILE_H;

    __shared__ float lds[LDS_ROWS * LDS_W];

    const float* __restrict__ xplane = xin  + (size_t)plane * PLANE;
    const float* __restrict__ wplane = wgt  + (size_t)plane * 9 * PLANE;
    float*       __restrict__ oplane = xout + (size_t)plane * PLANE;

    // ---- Stage haloed x tile into LDS via CDNA5 async global->LDS b128 DMA ----
    // 18 rows x 32 float4 quads; index math is shifts/masks only.
    for (int idx = tid; idx < LDS_ROWS * 32; idx += NTHREADS) {
        const int r  = idx >> 5;            // 0..17
        const int q  = idx & 31;            // quad within row
        const int gh = row0 - 1 + r;
        const int l  = r * LDS_W + 4 + q * 4;   // byte addr r*544 + 16*(q+1): 16B aligned
        if (gh >= 0 && gh < H) {
            const float* gp = xplane + gh * W + q * 4;   // 16B aligned
            unsigned lds_addr = (unsigned)(uintptr_t)(&lds[l]);
            asm volatile("global_load_async_to_lds_b128 %0, %1, off"
                         :: "v"(lds_addr), "v"(gp)
                         : "memory");
        } else {
            float4 z = make_float4(0.f, 0.f, 0.f, 0.f);
            *(float4*)(&lds[l]) = z;        // zero-pad OOB halo row
        }
    }
    // Halo columns (gw = -1 and gw = 128) are outside the image for every tile: zero once.
    if (tid < 2 * LDS_ROWS) {
        const int r = tid >> 1;
        const int c = (tid & 1) ? (LDS_W - 4) : 3;   // 132 : 3
        lds[r * LDS_W + c] = 0.0f;
    }
    asm volatile("s_wait_asynccnt 0x0" ::: "memory");
    __syncthreads();

    // ---- Compute: each thread owns a 4-wide pixel quad, 2 row iterations ----
    const int tx  = tid & 31;   // quad column index
    const int tyq = tid >> 5;   // 0..7
    const int w0  = tx * 4;

    for (int rr = tyq; rr < TILE_H; rr += 8) {
        const int gh = row0 + rr;

        // Prefetch the next row-chunk's weight lines (streamed data).
        if (rr + 8 < TILE_H) {
#pragma unroll
            for (int k = 0; k < 9; ++k)
                __builtin_prefetch(wplane + k * PLANE + (gh + 8) * W + w0, 0, 0);
        }

        // 9 coalesced b128 weight loads, abs + tap-sum for normalization.
        float4 wv[9];
        float sx = 0.f, sy = 0.f, sz = 0.f, sw = 0.f;
#pragma unroll
        for (int k = 0; k < 9; ++k) {
            float4 v = *(const float4*)(wplane + k * PLANE + gh * W + w0);
            v.x = fabsf(v.x); v.y = fabsf(v.y); v.z = fabsf(v.z); v.w = fabsf(v.w);
            sx += v.x; sy += v.y; sz += v.z; sw += v.w;
            wv[k] = v;
        }

        float ax = 0.f, ay = 0.f, az = 0.f, aw = 0.f;
#pragma unroll
        for (int ki = 0; ki < 3; ++ki) {
            const float* lr = &lds[(rr + ki) * LDS_W + w0 + 3];
            float xr[6];
#pragma unroll
            for (int j = 0; j < 6; ++j) xr[j] = lr[j];
#pragma unroll
            for (int kj = 0; kj < 3; ++kj) {
                const float4 wk = wv[ki * 3 + kj];
                ax = fmaf(wk.x, xr[kj + 0], ax);
                ay = fmaf(wk.y, xr[kj + 1], ay);
                az = fmaf(wk.z, xr[kj + 2], az);
                aw = fmaf(wk.w, xr[kj + 3], aw);
            }
        }

        float4 o;
        o.x = ax / sx; o.y = ay / sy; o.z = az / sz; o.w = aw / sw;
        *(float4*)(oplane + gh * W + w0) = o;
    }
}

extern "C" void kernel_launch(void* const* d_in, const int* in_sizes, int n_in,
                              void* d_out, int out_size, void* d_ws, size_t ws_size,
                              hipStream_t stream)
{
    const float* x   = (const float*)d_in[0];
    const float* w   = (const float*)d_in[1];
    float*       out = (float*)d_out;
    float*       tmp = (float*)d_ws;   // 16 MB ping buffer

    const int planes = in_sizes[0] / PLANE;   // n*c = 256
    const dim3 grid(planes * TPP);            // 2048 blocks
    const dim3 block(NTHREADS);

    // steps = min(max_step=8, max(h,w)=128) = 8 (static in the reference)
    const int steps = 8;
    const float* src = x;
    for (int s = 0; s < steps; ++s) {
        float* dst = ((s & 1) == 0) ? tmp : out;   // even step count => final lands in d_out
        diffusion_step<<<grid, block, 0, stream>>>(src, w, dst);
        src = dst;
    }
}